// FullyConnectedSeparableFiberBundleConv_19602230739215
// MI455X (gfx1250) — compile-verified
//
#include <hip/hip_runtime.h>
#include <hip/hip_bf16.h>

// Problem constants (from reference): B=1, N=M=256, O=12, C=128, D=32
#define MPTS 256
#define NPTS 256
#define ORI  12
#define CH   128
#define BD   32

typedef float v2f __attribute__((ext_vector_type(2)));
typedef float v8f __attribute__((ext_vector_type(8)));

// ---------------------------------------------------------------------------
// Kernel 1: rot_kernel[p,o,c] = sum_d fkb[p,o,d] * W_rot[d,c]   (12*12*128)
// ---------------------------------------------------------------------------
__global__ __launch_bounds__(128) void rot_proj_kernel(
    const float* __restrict__ fkb,   // [O,O,D]
    const float* __restrict__ Wr,    // [D,C]
    float* __restrict__ rot)         // [O,O,C]
{
    const int idx = blockIdx.x * blockDim.x + threadIdx.x;   // 0 .. 18431
    if (idx >= ORI * ORI * CH) return;
    const int c  = idx & (CH - 1);
    const int po = idx >> 7;                                 // p*12 + o
    float s = 0.0f;
#pragma unroll
    for (int d = 0; d < BD; ++d)
        s += fkb[po * BD + d] * Wr[d * CH + c];
    rot[idx] = s;
}

// ---------------------------------------------------------------------------
// Kernel 2: fused spatial GEMM per orientation, f32 WMMA 16x16x4.
//   y[m,o,c] = sum_{n,d} kb[m,n,o,d] * (W[d,c] * x[n,o,c] * mask[n])
// Grid: 12 * (256/16) = 192 blocks of 256 threads (8 waves).
// Block => (o, 16-row m-tile). Wave w => 16 output columns c0 = 16*w.
// A (16 x K) staged in LDS (shared by all 8 waves), software-pipelined with
// register prefetch of the next stage. B built in registers (rank-structured:
// B[(n,d),c] = W[d,c]*x[n,o,c]*mask[n]). Two accumulators break the serial
// WMMA D->C dependency chain.
// ---------------------------------------------------------------------------
__global__ __launch_bounds__(256) void spatial_gemm_kernel(
    const float* __restrict__ x,     // [N,O,C]
    const float* __restrict__ kb,    // [M,N,O,D]
    const float* __restrict__ mask,  // [N]
    const float* __restrict__ W,     // [D,C]
    float* __restrict__ y)           // [M,O,C]
{
    // 16 rows x 128 K-values per stage; row stride 132 => LDS banks
    // (4*m + k) mod 64 are all distinct across the 16 rows (conflict-free).
    __shared__ float ldsA[16 * 132];

    const int o  = blockIdx.x % ORI;
    const int M0 = (blockIdx.x / ORI) * 16;

    const int t    = threadIdx.x;
    const int wave = t >> 5;          // 0..7
    const int lane = t & 31;
    const int cl   = lane & 15;       // output column within wave tile / A row
    const int hi   = lane >> 4;       // half-wave: K-pair select for A/B frags
    const int c0   = wave * 16;

    // Per-lane W values actually needed by this half-wave:
    // B frag rows for step j (K-chunk kk=4j) are d = 4j+2*hi and 4j+2*hi+1.
    float wa[8], wb[8];
#pragma unroll
    for (int j = 0; j < 8; ++j) {
        const int d0 = 4 * j + 2 * hi;
        wa[j] = W[d0 * CH + c0 + cl];
        wb[j] = W[(d0 + 1) * CH + c0 + cl];
    }

    // Cooperative A-load mapping: thread -> (row mr, point nn_ld, 8-float run)
    const int mr    = t >> 4;         // 0..15
    const int sub   = t & 15;
    const int nn_ld = sub >> 2;       // 0..3
    const int d8    = (sub & 3) << 3; // 0,8,16,24

    const float* src0 = kb + (size_t)(M0 + mr) * (NPTS * ORI * BD)
                           + (size_t)nn_ld * (ORI * BD)
                           + o * BD + d8;
    const size_t stage_stride = (size_t)4 * (ORI * BD);  // 4 points per stage
    float* dst = &ldsA[mr * 132 + nn_ld * 32 + d8];

    v8f acc0 = {};
    v8f acc1 = {};

    // ---- prologue: prefetch stage 0 into registers ----
    float4 r0 = *(const float4*)(src0);
    float4 r1 = *(const float4*)(src0 + 4);

    for (int nb = 0; nb < NPTS; nb += 4) {
        __syncthreads();               // previous stage fully consumed
        *(float4*)(dst)     = r0;      // commit current stage to LDS
        *(float4*)(dst + 4) = r1;

        // prefetch next stage while this one is computed
        if (nb + 4 < NPTS) {
            const float* nsrc = src0 + (size_t)(nb + 4) / 4 * stage_stride;
            r0 = *(const float4*)(nsrc);
            r1 = *(const float4*)(nsrc + 4);
        }
        __syncthreads();               // LDS tile visible to all waves

        // ---- consume: 4 points x 8 K-steps of WMMA f32 16x16x4 ----
#pragma unroll
        for (int nn = 0; nn < 4; ++nn) {
            const int n = nb + nn;
            const float xv = x[(size_t)n * (ORI * CH) + o * CH + c0 + cl]
                             * mask[n];
#pragma unroll
            for (int j = 0; j < 8; ++j) {
                const int kk = 4 * j;
                // A frag (16x4, f32): lane holds A[m=cl, kk+2*hi .. +1]
                const float* ap = &ldsA[cl * 132 + nn * 32 + kk + 2 * hi];
                v2f a; a.x = ap[0]; a.y = ap[1];
                // B frag (4x16): lane holds B[kk+2*hi .. +1, n=cl]
                v2f b; b.x = wa[j] * xv; b.y = wb[j] * xv;
                if (j & 1) {
                    acc1 = __builtin_amdgcn_wmma_f32_16x16x4_f32(
                        false, a, false, b, (short)0, acc1, false, false);
                } else {
                    acc0 = __builtin_amdgcn_wmma_f32_16x16x4_f32(
                        false, a, false, b, (short)0, acc0, false, false);
                }
            }
        }
    }

    const v8f acc = acc0 + acc1;

    // C/D layout: VGPR r holds (M=r, N=cl) for lanes 0-15, (M=r+8) for 16-31.
#pragma unroll
    for (int r = 0; r < 8; ++r) {
        const int m = M0 + r + 8 * hi;
        y[(size_t)m * (ORI * CH) + o * CH + c0 + cl] = acc[r];
    }
}

// ---------------------------------------------------------------------------
// Kernel 3: out[m,p,c] = bias[c] + (1/12) * sum_o y[m,o,c] * rot[p,o,c]
// ---------------------------------------------------------------------------
__global__ __launch_bounds__(256) void fiber_rot_kernel(
    const float* __restrict__ y,     // [M,O,C]
    const float* __restrict__ rot,   // [O,O,C]
    const float* __restrict__ bias,  // [C]
    float* __restrict__ out)         // [M,O,C]  (p in place of o)
{
    const int idx = blockIdx.x * blockDim.x + threadIdx.x;
    if (idx >= MPTS * ORI * CH) return;
    const int c = idx & (CH - 1);
    const int p = (idx >> 7) % ORI;
    const int m = idx / (ORI * CH);
    float s = 0.0f;
#pragma unroll
    for (int o = 0; o < ORI; ++o)
        s += y[(size_t)(m * ORI + o) * CH + c] * rot[(p * ORI + o) * CH + c];
    out[idx] = s * (1.0f / (float)ORI) + bias[c];
}

// ---------------------------------------------------------------------------
extern "C" void kernel_launch(void* const* d_in, const int* in_sizes, int n_in,
                              void* d_out, int out_size, void* d_ws, size_t ws_size,
                              hipStream_t stream) {
    const float* x    = (const float*)d_in[0];  // (1,256,12,128)
    const float* kb   = (const float*)d_in[1];  // (1,256,256,12,32)
    const float* fkb  = (const float*)d_in[2];  // (12,12,32)
    const float* mask = (const float*)d_in[3];  // (1,256)
    const float* Ws   = (const float*)d_in[4];  // (32,128)
    const float* Wr   = (const float*)d_in[5];  // (32,128)
    const float* bias = (const float*)d_in[6];  // (128,)
    float* out = (float*)d_out;                 // (1,256,12,128)

    float* rot  = (float*)d_ws;                 // 18432 f32 (pad to 32768)
    float* yint = rot + 32768;                  // 393216 f32

    rot_proj_kernel<<<(ORI * ORI * CH + 127) / 128, 128, 0, stream>>>(fkb, Wr, rot);
    spatial_gemm_kernel<<<ORI * (MPTS / 16), 256, 0, stream>>>(x, kb, mask, Ws, yint);
    fiber_rot_kernel<<<(MPTS * ORI * CH + 255) / 256, 256, 0, stream>>>(yint, rot, bias, out);
}